// LightConv_15126874816536
// MI455X (gfx1250) — compile-verified
//
#include <hip/hip_runtime.h>

// LightConv as banded GEMM on the CDNA5 matrix pipe, full fp32 via
// V_WMMA_F32_16X16X4_F32. One wave per (b, t-tile16, h); 12 chained WMMAs
// per 16x16 output tile, 4 s-tiles per head.

typedef __attribute__((ext_vector_type(2))) float v2f;
typedef __attribute__((ext_vector_type(8))) float v8f;

#define NB   8
#define NT   1024
#define NH   8
#define NS   64
#define NK   31
#define NC   512          // NH * NS
#define PADK 15           // NK / 2
#define NJ   12           // ceil((16 + NK - 1) / 4) -> band width 46 padded to 48

__global__ __launch_bounds__(256) void lightconv_wmma_f32(
    const float* __restrict__ x,        // (B, T, C)
    const float* __restrict__ filters,  // (B, T, H*K)
    const float* __restrict__ bias,     // (C)
    float* __restrict__ out)            // (B, T, C)
{
    // Per-wave softmax-weight scratch: [wave][t-row m][k], k padded 31->32.
    __shared__ float wsm[8][16][32];

    const int lane = threadIdx.x & 31;
    const int wave = threadIdx.x >> 5;
    const int task = blockIdx.x * 8 + wave;   // 4096 tasks total, no remainder
    const int h    = task & 7;
    const int tt   = (task >> 3) & 63;
    const int b    = task >> 9;
    const int t0   = tt * 16;

    // ---- softmax over K for the 16 t-rows of this tile (lanes 0..15) ----
    const int m = lane & 15;
    if (lane < 16) {
        const float* f = filters + (size_t)(b * NT + t0 + m) * (NH * NK) + h * NK;
        float v[NK];
        float mx = -3.4e38f;
#pragma unroll
        for (int k = 0; k < NK; ++k) { v[k] = f[k]; mx = fmaxf(mx, v[k]); }
        float sum = 0.f;
#pragma unroll
        for (int k = 0; k < NK; ++k) { v[k] = __expf(v[k] - mx); sum += v[k]; }
        const float inv = 1.f / sum;
#pragma unroll
        for (int k = 0; k < NK; ++k) wsm[wave][m][k] = v[k] * inv;
    }
    __syncthreads();   // wave-local producer/consumer; also keeps EXEC uniform for WMMA

    // ---- build banded A fragments (16x4 f32 layout: V0 holds K=0/2, V1 K=1/3) ----
    const int klo = (lane < 16) ? 0 : 2;
    v2f afrag[NJ];
#pragma unroll
    for (int j = 0; j < NJ; ++j) {
        const int jj0 = 4 * j + klo;      // column in the 16x48 banded matrix
        const int k0  = jj0 - m;          // filter tap index
        const int k1  = k0 + 1;
        afrag[j].x = (k0 >= 0 && k0 < NK) ? wsm[wave][m][k0] : 0.f;
        afrag[j].y = (k1 >= 0 && k1 < NK) ? wsm[wave][m][k1] : 0.f;
    }

    // ---- 4 s-tiles of 16 columns: load B slabs, chain 12 WMMAs, store + bias ----
    const int n     = lane & 15;
    const int rbase = t0 - PADK + klo;    // first x row for this lane's B element
    for (int st = 0; st < 4; ++st) {
        const int c0 = h * NS + st * 16;
        const float* xcol = x + (size_t)b * NT * NC + c0 + n;

        v8f acc = {};
#pragma unroll
        for (int j = 0; j < NJ; ++j) {
            const int r0 = rbase + 4 * j;
            const int r1 = r0 + 1;
            v2f bfrag;
            bfrag.x = (r0 >= 0 && r0 < NT) ? xcol[(size_t)r0 * NC] : 0.f;
            bfrag.y = (r1 >= 0 && r1 < NT) ? xcol[(size_t)r1 * NC] : 0.f;
            // D = A(16x4) * B(4x16) + C   (full fp32)
            acc = __builtin_amdgcn_wmma_f32_16x16x4_f32(
                      false, afrag[j], false, bfrag, (short)0, acc, false, false);
        }

        const float bv    = bias[c0 + n];
        const int   mrow0 = (lane < 16) ? 0 : 8;          // C/D layout: V_i -> rows i, i+8
        float* orow = out + (size_t)(b * NT + t0 + mrow0) * NC + c0 + n;
#pragma unroll
        for (int vi = 0; vi < 8; ++vi)
            orow[(size_t)vi * NC] = acc[vi] + bv;
    }
}

extern "C" void kernel_launch(void* const* d_in, const int* in_sizes, int n_in,
                              void* d_out, int out_size, void* d_ws, size_t ws_size,
                              hipStream_t stream) {
    (void)in_sizes; (void)n_in; (void)out_size; (void)d_ws; (void)ws_size;
    const float* x       = (const float*)d_in[0];
    const float* filters = (const float*)d_in[1];
    const float* bias    = (const float*)d_in[2];
    float*       out     = (float*)d_out;

    // 4096 wave-tasks = (B=8) * (T/16=64) * (H=8); 8 waves per 256-thread block.
    dim3 grid(512), block(256);
    lightconv_wmma_f32<<<grid, block, 0, stream>>>(x, filters, bias, out);
}